// GConv_86998857548306
// MI455X (gfx1250) — compile-verified
//
#include <hip/hip_runtime.h>

#define N_NODES 50000
#define N_EDGES 800000
#define DIM     128
#define NLAYER  3
#define NGRAPH  512
#define ZSTRIDE (NLAYER * DIM)   // 384
#define BN_EPS  1e-5f
#define LDS_STRIDE 132           // 128 + 4 pad: m*132 % 64 distinct -> conflict-free

typedef __attribute__((ext_vector_type(2))) float v2f;
typedef __attribute__((ext_vector_type(8))) float v8f;

// ---------------------------------------------------------------------------
// 1) GIN aggregation: agg[dst] += z[src] over all edges, 4 feats per thread.
// ---------------------------------------------------------------------------
__global__ __launch_bounds__(256) void scatter_kernel(
    const float* __restrict__ z, int zstride,
    const int* __restrict__ src, const int* __restrict__ dst,
    float* __restrict__ agg)
{
    int tid = blockIdx.x * blockDim.x + threadIdx.x;
    if (tid >= N_EDGES * 32) return;
    int e  = tid >> 5;
    int c  = (tid & 31) * 4;
    int s  = src[e];
    int d  = dst[e];
    float4 v = *(const float4*)(z + (size_t)s * zstride + c);
    float* a = agg + (size_t)d * DIM + c;
    atomicAdd(a + 0, v.x);
    atomicAdd(a + 1, v.y);
    atomicAdd(a + 2, v.z);
    atomicAdd(a + 3, v.w);
}

// ---------------------------------------------------------------------------
// 2) Fused GIN MLP: h = relu(relu((z+agg)@W1 + b1)@W2 + b2), fp32 WMMA.
//    Block = 8 waves, owns 16 rows; wave w owns output cols [16w,16w+16).
//    hbuf is read as agg at block entry and overwritten with h at exit
//    (same rows only -> safe aliasing). Also accumulates BN sum/sumsq.
// ---------------------------------------------------------------------------
__global__ __launch_bounds__(256) void gin_mlp_kernel(
    const float* __restrict__ z, int zstride,
    float* __restrict__ hbuf,
    const float* __restrict__ W1, const float* __restrict__ b1,
    const float* __restrict__ W2, const float* __restrict__ b2,
    float* __restrict__ stats)
{
    __shared__ float ldsA[16 * LDS_STRIDE];
    __shared__ float ldsH[16 * LDS_STRIDE];

    const int tid  = threadIdx.x;
    const int wave = tid >> 5;
    const int lane = tid & 31;
    const int half = lane >> 4;       // 0: K lo pair / M 0..7 ; 1: K hi pair / M 8..15
    const int l16  = lane & 15;
    const int row0 = blockIdx.x * 16; // N = 3125*16 exactly, no tail
    const int col  = wave * 16 + l16;

    // Stage A = z + agg into LDS (16 x 128, padded)
    for (int idx = tid; idx < 16 * DIM; idx += 256) {
        int r = idx >> 7, c = idx & 127;
        ldsA[r * LDS_STRIDE + c] =
            z[(size_t)(row0 + r) * zstride + c] + hbuf[(size_t)(row0 + r) * DIM + c];
    }
    __syncthreads();

    // GEMM1: acc = A @ W1 via v_wmma_f32_16x16x4_f32, K = 128 in steps of 4
    v8f acc = {};
    for (int k0 = 0; k0 < DIM; k0 += 4) {
        int ka = k0 + 2 * half;
        v2f a, b;
        a.x = ldsA[l16 * LDS_STRIDE + ka];
        a.y = ldsA[l16 * LDS_STRIDE + ka + 1];
        b.x = W1[ka * DIM + col];
        b.y = W1[(ka + 1) * DIM + col];
        acc = __builtin_amdgcn_wmma_f32_16x16x4_f32(
            false, a, false, b, (short)0, acc, false, false);
    }

    // bias + relu -> stage H1 in LDS
    float bias1 = b1[col];
    #pragma unroll
    for (int g = 0; g < 8; ++g) {
        int m = g + 8 * half;
        float v = acc[g] + bias1;
        ldsH[m * LDS_STRIDE + col] = v > 0.f ? v : 0.f;
    }
    __syncthreads();

    // GEMM2: acc2 = H1 @ W2
    v8f acc2 = {};
    for (int k0 = 0; k0 < DIM; k0 += 4) {
        int ka = k0 + 2 * half;
        v2f a, b;
        a.x = ldsH[l16 * LDS_STRIDE + ka];
        a.y = ldsH[l16 * LDS_STRIDE + ka + 1];
        b.x = W2[ka * DIM + col];
        b.y = W2[(ka + 1) * DIM + col];
        acc2 = __builtin_amdgcn_wmma_f32_16x16x4_f32(
            false, a, false, b, (short)0, acc2, false, false);
    }

    // bias + relu, store h, accumulate BN column stats
    float bias2 = b2[col];
    float s = 0.f, q = 0.f;
    #pragma unroll
    for (int g = 0; g < 8; ++g) {
        int m = g + 8 * half;
        float v = acc2[g] + bias2;
        v = v > 0.f ? v : 0.f;
        hbuf[(size_t)(row0 + m) * DIM + col] = v;
        s += v;
        q += v * v;
    }
    // lanes l and l^16 hold the same column: combine, then one atomic per column
    s += __shfl_xor(s, 16);
    q += __shfl_xor(q, 16);
    if (half == 0) {
        atomicAdd(&stats[col], s);
        atomicAdd(&stats[DIM + col], q);
    }
}

// ---------------------------------------------------------------------------
// 3) Finalize BN stats: stats[c] <- mu, stats[D+c] <- rsqrt(var + eps)
// ---------------------------------------------------------------------------
__global__ void bn_stats_finalize(float* __restrict__ stats)
{
    int c = threadIdx.x;
    float mu  = stats[c] * (1.0f / N_NODES);
    float var = stats[DIM + c] * (1.0f / N_NODES) - mu * mu;
    stats[c]       = mu;
    stats[DIM + c] = rsqrtf(var + BN_EPS);
}

// ---------------------------------------------------------------------------
// 4) Apply BN, write z_cat slice (next layer input), pool into g_cat slice.
// ---------------------------------------------------------------------------
__global__ __launch_bounds__(256) void bn_pool_kernel(
    const float* __restrict__ h,
    const float* __restrict__ stats,
    const float* __restrict__ gamma, const float* __restrict__ beta,
    const int* __restrict__ batch,
    float* __restrict__ zcat, float* __restrict__ gcat)
{
    int tid = blockIdx.x * blockDim.x + threadIdx.x;
    if (tid >= N_NODES * 32) return;
    int i = tid >> 5;
    int c = (tid & 31) * 4;
    int g = batch[i];
    float4 v = *(const float4*)(h + (size_t)i * DIM + c);
    float in[4] = {v.x, v.y, v.z, v.w};
    float4 o;
    float* op = &o.x;
    #pragma unroll
    for (int j = 0; j < 4; ++j) {
        float mu = stats[c + j];
        float rs = stats[DIM + c + j];
        float y  = (in[j] - mu) * rs * gamma[c + j] + beta[c + j];
        op[j] = y;
        atomicAdd(&gcat[(size_t)g * ZSTRIDE + c + j], y);
    }
    *(float4*)(zcat + (size_t)i * ZSTRIDE + c) = o;
}

// ---------------------------------------------------------------------------
extern "C" void kernel_launch(void* const* d_in, const int* in_sizes, int n_in,
                              void* d_out, int out_size, void* d_ws, size_t ws_size,
                              hipStream_t stream)
{
    (void)in_sizes; (void)n_in; (void)out_size; (void)ws_size;

    const float* x     = (const float*)d_in[0];
    const int*   ei    = (const int*)  d_in[1];
    const int*   batch = (const int*)  d_in[2];
    const float* W1    = (const float*)d_in[3];
    const float* b1    = (const float*)d_in[4];
    const float* W2    = (const float*)d_in[5];
    const float* b2    = (const float*)d_in[6];
    const float* gamma = (const float*)d_in[7];
    const float* beta  = (const float*)d_in[8];

    float* out  = (float*)d_out;
    float* zcat = out;                                    // [N, L*D]
    float* gcat = out + (size_t)N_NODES * ZSTRIDE;        // [G, L*D]

    float* agg   = (float*)d_ws;                          // N*D floats (aliased h)
    float* stats = agg + (size_t)N_NODES * DIM;           // 2*D floats

    const int* src = ei;             // edge_index[0]
    const int* dst = ei + N_EDGES;   // edge_index[1]

    hipMemsetAsync(gcat, 0, sizeof(float) * (size_t)NGRAPH * ZSTRIDE, stream);

    for (int l = 0; l < NLAYER; ++l) {
        const float* zin = (l == 0) ? x : (zcat + (size_t)(l - 1) * DIM);
        int zstride      = (l == 0) ? DIM : ZSTRIDE;

        hipMemsetAsync(agg,   0, sizeof(float) * (size_t)N_NODES * DIM, stream);
        hipMemsetAsync(stats, 0, sizeof(float) * 2 * DIM, stream);

        {
            long long threads = (long long)N_EDGES * 32;
            int blocks = (int)((threads + 255) / 256);
            scatter_kernel<<<blocks, 256, 0, stream>>>(zin, zstride, src, dst, agg);
        }

        gin_mlp_kernel<<<N_NODES / 16, 256, 0, stream>>>(
            zin, zstride, agg,
            W1 + (size_t)l * DIM * DIM, b1 + (size_t)l * DIM,
            W2 + (size_t)l * DIM * DIM, b2 + (size_t)l * DIM,
            stats);

        bn_stats_finalize<<<1, DIM, 0, stream>>>(stats);

        {
            long long threads = (long long)N_NODES * 32;
            int blocks = (int)((threads + 255) / 256);
            bn_pool_kernel<<<blocks, 256, 0, stream>>>(
                agg, stats,
                gamma + (size_t)l * DIM, beta + (size_t)l * DIM,
                batch,
                zcat + (size_t)l * DIM, gcat + (size_t)l * DIM);
        }
    }
}